// ErrorAwareSelfAttention_41729902248629
// MI455X (gfx1250) — compile-verified
//
#include <hip/hip_runtime.h>
#include <hip/hip_bf16.h>

// ---------------------------------------------------------------------------
// Error-aware windowed self-attention for MI455X (gfx1250), fp32 WMMA path,
// with CDNA5 async global->LDS tile staging (ASYNCcnt) + prefetch.
// ---------------------------------------------------------------------------

typedef __attribute__((ext_vector_type(2))) float v2f;
typedef __attribute__((ext_vector_type(8))) float v8f;

#define DIMC   512
#define NHEAD  4
#define CH     128
#define T_FR   4
#define HI     48
#define WIW    48
#define NPIX   (HI * WIW)        // 2304
#define LT     2
#define NWH    6
#define NWW    6
#define NWIN   36
#define WN     64
#define NROLL  192
#define NPOOL  144
#define NK     400               // WN + NROLL + NPOOL
#define NTOK   (T_FR * NPIX)     // 9216
#define ATT_SCALE 0.088388347648318447f  // 1/sqrt(128)

__device__ __forceinline__ v8f wmma_f32(v2f a, v2f b, v8f c) {
  // D = A(16x4) * B(4x16) + C(16x16), fp32
  return __builtin_amdgcn_wmma_f32_16x16x4_f32(
      /*neg_a=*/false, a, /*neg_b=*/false, b,
      /*c_mod=*/(short)0, c, /*reuse_a=*/false, /*reuse_b=*/false);
}

// Async 16B copy: global memory -> LDS, tracked by ASYNCcnt (no VGPR bounce).
__device__ __forceinline__ void async_copy_b128(void* lds, const void* gptr) {
  asm volatile("global_load_async_to_lds_b128 %0, %1, off"
               :: "v"((unsigned)(uintptr_t)lds), "v"(gptr)
               : "memory");
}
__device__ __forceinline__ void wait_async0() {
  asm volatile("s_wait_asynccnt 0x0" ::: "memory");
}

__device__ __forceinline__ float apply_act(float v, int act) {
  if (act == 1) return v > 0.f ? v : 0.2f * v;            // leaky_relu(0.2)
  if (act == 2) { float s = 1.f / (1.f + __expf(-v)); return v * s; } // SiLU
  return v;
}

// ---------------------------------------------------------------------------
// Generic fp32 WMMA GEMM: C[M,N] = act(A[M,K] @ B + bias)
//   transB==0: B is K x N row-major (ldb = N)
//   transB==1: B is N x K row-major (ldb = K)   (conv-style weights)
// Requires M%64==0, N%64==0, K%16==0. Block = 256 threads (8 waves).
// ---------------------------------------------------------------------------
__global__ __launch_bounds__(256)
void wmma_gemm_kernel(const float* __restrict__ A, const float* __restrict__ B,
                      const float* __restrict__ bias, float* __restrict__ C,
                      int M, int N, int K, int ldb, int transB, int act) {
  __shared__ __align__(16) float As[64][16];   // [m][k]
  __shared__ __align__(16) float Bs[16][64];   // [k][n]

  const int mBlk = blockIdx.x * 64;
  const int nBlk = blockIdx.y * 64;
  const int tid  = threadIdx.x;
  const int lane = tid & 31;
  const int wave = tid >> 5;
  const int tr   = wave >> 1;          // tile row 0..3
  const int tc0  = (wave & 1) * 2;     // tile cols {tc0, tc0+1}
  const int hl   = lane & 15;
  const int hi   = lane >> 4;          // 0: lanes 0-15, 1: lanes 16-31

  // per-thread staging coordinates
  const int ia = tid * 4;
  const int am = ia >> 4;              // A row in tile
  const int ak = ia & 15;              // A col in tile
  const int bk = ia >> 6;              // B row in tile
  const int bn = ia & 63;              // B col in tile

  v8f acc0 = (v8f)0.0f;
  v8f acc1 = (v8f)0.0f;

  for (int k0 = 0; k0 < K; k0 += 16) {
    __syncthreads();   // previous iteration's LDS reads complete
    // stage A tile (64x16) via async global->LDS
    async_copy_b128(&As[am][ak], A + (size_t)(mBlk + am) * K + k0 + ak);
    // stage B tile (16x64)
    if (!transB) {
      async_copy_b128(&Bs[bk][bn], B + (size_t)(k0 + bk) * ldb + nBlk + bn);
    } else {
      float4 vv;
      vv.x = B[(size_t)(nBlk + bn + 0) * ldb + k0 + bk];
      vv.y = B[(size_t)(nBlk + bn + 1) * ldb + k0 + bk];
      vv.z = B[(size_t)(nBlk + bn + 2) * ldb + k0 + bk];
      vv.w = B[(size_t)(nBlk + bn + 3) * ldb + k0 + bk];
      *(float4*)&Bs[bk][bn] = vv;
    }
    // warm L2->WGP for the next K block while WMMAs run
    if (k0 + 16 < K) {
      __builtin_prefetch(A + (size_t)(mBlk + am) * K + k0 + 16 + ak, 0, 3);
      if (!transB)
        __builtin_prefetch(B + (size_t)(k0 + 16 + bk) * ldb + nBlk + bn, 0, 3);
    }
    wait_async0();
    __syncthreads();

    // A frag: lane half 'hi' holds K = kk + hi*2 + {0,1} of row (tr*16 + hl)
    // B frag: lane half 'hi' holds K = kk + hi*2 + {0,1} of col (tc*16 + hl)
    for (int kk = 0; kk < 16; kk += 4) {
      v2f a, b0, b1;
      a.x  = As[tr * 16 + hl][kk + hi * 2 + 0];
      a.y  = As[tr * 16 + hl][kk + hi * 2 + 1];
      b0.x = Bs[kk + hi * 2 + 0][tc0 * 16 + hl];
      b0.y = Bs[kk + hi * 2 + 1][tc0 * 16 + hl];
      b1.x = Bs[kk + hi * 2 + 0][(tc0 + 1) * 16 + hl];
      b1.y = Bs[kk + hi * 2 + 1][(tc0 + 1) * 16 + hl];
      acc0 = wmma_f32(a, b0, acc0);
      acc1 = wmma_f32(a, b1, acc1);
    }
  }

  // Epilogue. D layout: VGPR r -> row (tr*16 + r + hi*8), col (tc*16 + hl).
  for (int r = 0; r < 8; ++r) {
    int m  = mBlk + tr * 16 + r + hi * 8;
    int n0 = nBlk + tc0 * 16 + hl;
    int n1 = n0 + 16;
    float v0 = apply_act(acc0[r] + bias[n0], act);
    float v1 = apply_act(acc1[r] + bias[n1], act);
    C[(size_t)m * N + n0] = v0;
    C[(size_t)m * N + n1] = v1;
  }
}

// ---------------------------------------------------------------------------
// dst[f,pix,c] = src[f,pix,c] + mask[f,pix]   (first LT frames of k or v)
// ---------------------------------------------------------------------------
__global__ __launch_bounds__(256)
void addmask_kernel(const float* __restrict__ src, const float* __restrict__ mask,
                    float* __restrict__ dst) {
  int idx = blockIdx.x * 256 + threadIdx.x;
  if (idx >= LT * NPIX * DIMC) return;
  int pf = idx >> 9;   // f*NPIX + pix
  dst[idx] = src[idx] + mask[pf];
}

// ---------------------------------------------------------------------------
// 3x3 conv 128->128, pad 1, over (LT, 48, 48), channel-last layout, + SiLU.
// gridDim.x = LT*128 (one (frame, oc) plane per block).
// ---------------------------------------------------------------------------
__global__ __launch_bounds__(256)
void conv3x3_silu_kernel(const float* __restrict__ in, const float* __restrict__ w,
                         const float* __restrict__ bias, float* __restrict__ out) {
  int oc = blockIdx.x & 127;
  int fr = blockIdx.x >> 7;
  __shared__ float wl[128 * 9];
  for (int i = threadIdx.x; i < 128 * 9; i += 256)
    wl[i] = w[(size_t)oc * 128 * 9 + i];
  __syncthreads();

  for (int pix = threadIdx.x; pix < NPIX; pix += 256) {
    int yy = pix / WIW, xx = pix % WIW;
    float acc = bias[oc];
    for (int dy = 0; dy < 3; ++dy) {
      int sy = yy + dy - 1;
      if ((unsigned)sy >= (unsigned)HI) continue;
      for (int dx = 0; dx < 3; ++dx) {
        int sx = xx + dx - 1;
        if ((unsigned)sx >= (unsigned)WIW) continue;
        const float* ip = in + (size_t)(fr * NPIX + sy * WIW + sx) * 128;
        const float* wp = wl + dy * 3 + dx;
        for (int ic = 0; ic < 128; ++ic)
          acc = fmaf(ip[ic], wp[ic * 9], acc);
      }
    }
    float s = 1.f / (1.f + __expf(-acc));
    out[(size_t)(fr * NPIX + pix) * 128 + oc] = acc * s;
  }
}

// ---------------------------------------------------------------------------
// Depthwise 4x4 stride-4 pooling conv over x: (T,48,48,512) -> px (T*144, 512)
// ---------------------------------------------------------------------------
__global__ __launch_bounds__(256)
void pool_kernel(const float* __restrict__ x, const float* __restrict__ pw,
                 const float* __restrict__ pb, float* __restrict__ px) {
  int idx = blockIdx.x * 256 + threadIdx.x;
  if (idx >= T_FR * NPOOL * DIMC) return;
  int c = idx & 511;
  int rest = idx >> 9;
  int pp = rest % NPOOL;
  int t  = rest / NPOOL;
  int py = pp / 12, pxx = pp % 12;
  float acc = pb[c];
  for (int dy = 0; dy < 4; ++dy)
    for (int dx = 0; dx < 4; ++dx)
      acc = fmaf(x[(size_t)(t * NPIX + (py * 4 + dy) * WIW + pxx * 4 + dx) * DIMC + c],
                 pw[c * 16 + dy * 4 + dx], acc);
  px[(size_t)(t * NPOOL + pp) * DIMC + c] = acc;
}

// ---------------------------------------------------------------------------
// Per-window masked flag: any mask pixel > 0 over LT frames in the window.
// ---------------------------------------------------------------------------
__global__ __launch_bounds__(64)
void flags_kernel(const float* __restrict__ mask, int* __restrict__ flags) {
  __shared__ int f;
  int wi = blockIdx.x;
  if (threadIdx.x == 0) f = 0;
  __syncthreads();
  int wy = wi / NWW, wx = wi % NWW;
  for (int it = threadIdx.x; it < LT * WN; it += 64) {
    int fr = it >> 6, p = it & 63;
    int pix = (wy * 8 + (p >> 3)) * WIW + wx * 8 + (p & 7);
    if (mask[fr * NPIX + pix] > 0.f) atomicOr(&f, 1);
  }
  __syncthreads();
  if (threadIdx.x == 0) flags[wi] = f;
}

// ---------------------------------------------------------------------------
// Per-window key/value gather table: 400 entries.
//   idx < 2304  : pixel index within a frame of k/v
//   idx >= 2304 : pooled token (idx-2304) within a frame of pk/pv
// Order: own 64 pixels, then 4 rolls pruned by VALID_IND, then 144 pooled.
// ---------------------------------------------------------------------------
__global__ __launch_bounds__(64)
void idxtab_kernel(int* __restrict__ idxTab) {
  int wi = blockIdx.x * 64 + threadIdx.x;
  if (wi >= NWIN) return;
  int wy = wi / NWW, wx = wi % NWW;
  int* t = idxTab + wi * NK;
  int n = 0;
  for (int p = 0; p < WN; ++p)
    t[n++] = (wy * 8 + (p >> 3)) * WIW + wx * 8 + (p & 7);
  const int shy[4] = {-4, -4, 4, 4};
  const int shx[4] = {-4, 4, -4, 4};
  for (int r = 0; r < 4; ++r) {
    for (int p = 0; p < WN; ++p) {
      int py = p >> 3, pxx = p & 7;
      bool invalid;
      if (r == 0)      invalid = (py < 4 && pxx < 4);
      else if (r == 1) invalid = (py < 4 && pxx >= 4);
      else if (r == 2) invalid = (py >= 4 && pxx < 4);
      else             invalid = (py >= 4 && pxx >= 4);
      if (invalid) continue;
      int sy = ((wy * 8 + py) - shy[r] + HI) % HI;   // rolled[i] = src[i - shift]
      int sx = ((wx * 8 + pxx) - shx[r] + WIW) % WIW;
      t[n++] = sy * WIW + sx;
    }
  }
  for (int p = 0; p < NPOOL; ++p) t[n++] = NPIX + p;
}

// ---------------------------------------------------------------------------
// Masked-window global attention. Block = 1 wave (32), flash-style online
// softmax. grid = (16 query tiles, NHEAD, NWIN). Queries span t*WN = 256,
// keys span t*NK = 1600 (100 tiles of 16).
// ---------------------------------------------------------------------------
__global__ __launch_bounds__(32)
void attn_global_kernel(const float* __restrict__ q, const float* __restrict__ k,
                        const float* __restrict__ v, const float* __restrict__ pk,
                        const float* __restrict__ pv, const int* __restrict__ idxTab,
                        const int* __restrict__ flags, float* __restrict__ y) {
  int wi = blockIdx.z;
  if (!flags[wi]) return;
  int h  = blockIdx.y;
  int qt = blockIdx.x;
  int lane = threadIdx.x;
  int hl = lane & 15, hi = lane >> 4;
  int wy = wi / NWW, wx = wi % NWW;

  __shared__ __align__(16) float qs[16][CH];
  __shared__ __align__(16) float ks[16][CH];
  __shared__ __align__(16) float vs[16][CH];
  __shared__ __align__(16) float pt[16][16];

  for (int r = 0; r < 16; ++r) {
    int qq = qt * 16 + r;
    int tt = qq >> 6, p = qq & 63;
    int pix = (wy * 8 + (p >> 3)) * WIW + wx * 8 + (p & 7);
    const float* src = q + (size_t)(tt * NPIX + pix) * DIMC + h * CH;
    async_copy_b128(&qs[r][lane * 4], src + lane * 4);
  }

  v8f acc[8];
  for (int i = 0; i < 8; ++i) acc[i] = (v8f)0.0f;
  float mrow[8], lrow[8];
  for (int r = 0; r < 8; ++r) { mrow[r] = -1e30f; lrow[r] = 0.f; }

  for (int kt = 0; kt < (T_FR * NK) / 16; ++kt) {
    __syncthreads();
    for (int r = 0; r < 16; ++r) {
      int tok = kt * 16 + r;
      int tt = tok / NK;
      int j  = tok - tt * NK;
      int idx = idxTab[wi * NK + j];
      const float* kr; const float* vr;
      if (idx < NPIX) {
        kr = k  + (size_t)(tt * NPIX + idx) * DIMC + h * CH;
        vr = v  + (size_t)(tt * NPIX + idx) * DIMC + h * CH;
      } else {
        int pj = idx - NPIX;
        kr = pk + (size_t)(tt * NPOOL + pj) * DIMC + h * CH;
        vr = pv + (size_t)(tt * NPOOL + pj) * DIMC + h * CH;
      }
      async_copy_b128(&ks[r][lane * 4], kr + lane * 4);
      async_copy_b128(&vs[r][lane * 4], vr + lane * 4);
    }
    wait_async0();
    __syncthreads();

    // S = Q * K^T  (D layout: VGPR r -> query row r+hi*8, lane -> key col hl)
    v8f s = (v8f)0.0f;
    for (int kk = 0; kk < CH; kk += 4) {
      v2f a, b;
      a.x = qs[hl][kk + hi * 2 + 0];
      a.y = qs[hl][kk + hi * 2 + 1];
      b.x = ks[hl][kk + hi * 2 + 0];
      b.y = ks[hl][kk + hi * 2 + 1];
      s = wmma_f32(a, b, s);
    }

    float alpha[8];
    for (int r = 0; r < 8; ++r) {
      float sv = s[r] * ATT_SCALE;
      float mx = sv;
      for (int off = 1; off < 16; off <<= 1)
        mx = fmaxf(mx, __shfl_xor(mx, off, 32));
      float mnew = fmaxf(mrow[r], mx);
      float al = __expf(mrow[r] - mnew);
      float pe = __expf(sv - mnew);
      float rs = pe;
      for (int off = 1; off < 16; off <<= 1)
        rs += __shfl_xor(rs, off, 32);
      lrow[r] = lrow[r] * al + rs;
      mrow[r] = mnew;
      alpha[r] = al;
      pt[r + hi * 8][hl] = pe;      // store P row-major for transpose read
    }
    for (int dt = 0; dt < 8; ++dt)
      for (int r = 0; r < 8; ++r)
        acc[dt][r] *= alpha[r];
    __syncthreads();

    // O += P(16x16) * V(16x128)
    for (int kk = 0; kk < 16; kk += 4) {
      v2f a;
      a.x = pt[hl][kk + hi * 2 + 0];
      a.y = pt[hl][kk + hi * 2 + 1];
      for (int dt = 0; dt < 8; ++dt) {
        v2f b;
        b.x = vs[kk + hi * 2 + 0][dt * 16 + hl];
        b.y = vs[kk + hi * 2 + 1][dt * 16 + hl];
        acc[dt] = wmma_f32(a, b, acc[dt]);
      }
    }
  }

  for (int r = 0; r < 8; ++r) {
    int qq = qt * 16 + r + hi * 8;
    int tt = qq >> 6, p = qq & 63;
    int pix = (wy * 8 + (p >> 3)) * WIW + wx * 8 + (p & 7);
    float inv = 1.f / lrow[r];
    for (int dt = 0; dt < 8; ++dt)
      y[(size_t)(tt * NPIX + pix) * DIMC + h * CH + dt * 16 + hl] = acc[dt][r] * inv;
  }
}

// ---------------------------------------------------------------------------
// Unmasked-window local attention: per-frame 64x64 over own window tokens.
// grid = (16 = tt*4 + query-tile, NHEAD, NWIN), block = 1 wave.
// ---------------------------------------------------------------------------
__global__ __launch_bounds__(32)
void attn_local_kernel(const float* __restrict__ q, const float* __restrict__ k,
                       const float* __restrict__ v, const int* __restrict__ flags,
                       float* __restrict__ y) {
  int wi = blockIdx.z;
  if (flags[wi]) return;
  int h  = blockIdx.y;
  int tt = blockIdx.x >> 2;
  int qt = blockIdx.x & 3;
  int lane = threadIdx.x;
  int hl = lane & 15, hi = lane >> 4;
  int wy = wi / NWW, wx = wi % NWW;

  __shared__ __align__(16) float qs[16][CH];
  __shared__ __align__(16) float ks[16][CH];
  __shared__ __align__(16) float vs[16][CH];
  __shared__ __align__(16) float pt[16][16];

  for (int r = 0; r < 16; ++r) {
    int p = qt * 16 + r;
    int pix = (wy * 8 + (p >> 3)) * WIW + wx * 8 + (p & 7);
    const float* src = q + (size_t)(tt * NPIX + pix) * DIMC + h * CH;
    async_copy_b128(&qs[r][lane * 4], src + lane * 4);
  }

  v8f acc[8];
  for (int i = 0; i < 8; ++i) acc[i] = (v8f)0.0f;
  float mrow[8], lrow[8];
  for (int r = 0; r < 8; ++r) { mrow[r] = -1e30f; lrow[r] = 0.f; }

  for (int kt = 0; kt < 4; ++kt) {
    __syncthreads();
    for (int r = 0; r < 16; ++r) {
      int j = kt * 16 + r;
      int pix = (wy * 8 + (j >> 3)) * WIW + wx * 8 + (j & 7);
      const float* kr = k + (size_t)(tt * NPIX + pix) * DIMC + h * CH;
      const float* vr = v + (size_t)(tt * NPIX + pix) * DIMC + h * CH;
      async_copy_b128(&ks[r][lane * 4], kr + lane * 4);
      async_copy_b128(&vs[r][lane * 4], vr + lane * 4);
    }
    wait_async0();
    __syncthreads();

    v8f s = (v8f)0.0f;
    for (int kk = 0; kk < CH; kk += 4) {
      v2f a, b;
      a.x = qs[hl][kk + hi * 2 + 0];
      a.y = qs[hl][kk + hi * 2 + 1];
      b.x = ks[hl][kk + hi * 2 + 0];
      b.y = ks[hl][kk + hi * 2 + 1];
      s = wmma_f32(a, b, s);
    }

    float alpha[8];
    for (int r = 0; r < 8; ++r) {
      float sv = s[r] * ATT_SCALE;
      float mx = sv;
      for (int off = 1; off < 16; off <<= 1)
        mx = fmaxf(mx, __shfl_xor(mx, off, 32));
      float mnew = fmaxf(mrow[r], mx);
      float al = __expf(mrow[r] - mnew);
      float pe = __expf(sv - mnew);
      float rs = pe;
      for (int off = 1; off < 16; off <<= 1)
        rs += __shfl_xor(rs, off, 32);
      lrow[r] = lrow[r] * al + rs;
      mrow[r] = mnew;
      alpha[r] = al;
      pt[r + hi * 8][hl] = pe;
    }
    for (int dt = 0; dt < 8; ++dt)
      for (int r = 0; r < 8; ++r)
        acc[dt][r] *= alpha[r];
    __syncthreads();

    for (int kk = 0; kk < 16; kk += 4) {
      v2f a;
      a.x = pt[hl][kk + hi * 2 + 0];
      a.y = pt[hl][kk + hi * 2 + 1];
      for (int dt = 0; dt < 8; ++dt) {
        v2f b;
        b.x = vs[kk + hi * 2 + 0][dt * 16 + hl];
        b.y = vs[kk + hi * 2 + 1][dt * 16 + hl];
        acc[dt] = wmma_f32(a, b, acc[dt]);
      }
    }
  }

  for (int r = 0; r < 8; ++r) {
    int p = qt * 16 + r + hi * 8;
    int pix = (wy * 8 + (p >> 3)) * WIW + wx * 8 + (p & 7);
    float inv = 1.f / lrow[r];
    for (int dt = 0; dt < 8; ++dt)
      y[(size_t)(tt * NPIX + pix) * DIMC + h * CH + dt * 16 + hl] = acc[dt][r] * inv;
  }
}

// ---------------------------------------------------------------------------
extern "C" void kernel_launch(void* const* d_in, const int* in_sizes, int n_in,
                              void* d_out, int out_size, void* d_ws, size_t ws_size,
                              hipStream_t stream) {
  const float* x       = (const float*)d_in[0];
  const float* mask    = (const float*)d_in[1];
  const float* Wq      = (const float*)d_in[2];
  const float* bq      = (const float*)d_in[3];
  const float* Wk      = (const float*)d_in[4];
  const float* bk      = (const float*)d_in[5];
  const float* Wv      = (const float*)d_in[6];
  const float* bv      = (const float*)d_in[7];
  const float* Wp      = (const float*)d_in[8];
  const float* bp      = (const float*)d_in[9];
  const float* pool_w  = (const float*)d_in[10];
  const float* pool_b  = (const float*)d_in[11];
  const float* ksq_w   = (const float*)d_in[12];
  const float* ksq_b   = (const float*)d_in[13];
  const float* kf_w    = (const float*)d_in[14];
  const float* kf_b    = (const float*)d_in[15];
  const float* kun_w   = (const float*)d_in[16];
  const float* kun_b   = (const float*)d_in[17];
  const float* vsq_w   = (const float*)d_in[18];
  const float* vsq_b   = (const float*)d_in[19];
  const float* vf_w    = (const float*)d_in[20];
  const float* vf_b    = (const float*)d_in[21];
  const float* vun_w   = (const float*)d_in[22];
  const float* vun_b   = (const float*)d_in[23];
  float* out = (float*)d_out;

  // ---- workspace carve-up (floats) ----
  float* ws = (float*)d_ws;
  size_t o = 0;
  const size_t TOKSZ = (size_t)NTOK * DIMC;        // 4,718,592
  float* qb    = ws + o; o += TOKSZ;
  float* kb    = ws + o; o += TOKSZ;
  float* vb    = ws + o; o += TOKSZ;
  float* yb    = ws + o; o += TOKSZ;
  float* pxb   = ws + o; o += (size_t)T_FR * NPOOL * DIMC;  // 294,912
  float* pkb   = ws + o; o += (size_t)T_FR * NPOOL * DIMC;
  float* pvb   = ws + o; o += (size_t)T_FR * NPOOL * DIMC;
  float* modin = ws + o; o += (size_t)LT * NPIX * DIMC;     // 2,359,296
  float* t1    = ws + o; o += (size_t)LT * NPIX * 128;      // 589,824
  float* t2    = ws + o; o += (size_t)LT * NPIX * 128;
  int* idxTab  = (int*)(ws + o); o += NWIN * NK;            // 14,400
  int* flags   = (int*)(ws + o); o += 64;

  const dim3 blk256(256);
  const dim3 gemmFull(NTOK / 64, DIMC / 64);   // 144 x 8

  // q, k, v projections
  wmma_gemm_kernel<<<gemmFull, blk256, 0, stream>>>(x, Wq, bq, qb, NTOK, DIMC, DIMC, DIMC, 0, 0);
  wmma_gemm_kernel<<<gemmFull, blk256, 0, stream>>>(x, Wk, bk, kb, NTOK, DIMC, DIMC, DIMC, 0, 0);
  wmma_gemm_kernel<<<gemmFull, blk256, 0, stream>>>(x, Wv, bv, vb, NTOK, DIMC, DIMC, DIMC, 0, 0);

  const int nModElems = LT * NPIX * DIMC;
  const dim3 gAdd((nModElems + 255) / 256);
  const dim3 gSq(LT * NPIX / 64, 128 / 64);    // 72 x 2
  const dim3 gUn(LT * NPIX / 64, DIMC / 64);   // 72 x 8
  const dim3 gConv(LT * 128);

  // k modulator: (k[:2]+mask) -> 1x1 conv + leaky -> 3x3 conv + SiLU -> 1x1 conv
  addmask_kernel<<<gAdd, blk256, 0, stream>>>(kb, mask, modin);
  wmma_gemm_kernel<<<gSq, blk256, 0, stream>>>(modin, ksq_w, ksq_b, t1, LT * NPIX, 128, DIMC, DIMC, 1, 1);
  conv3x3_silu_kernel<<<gConv, blk256, 0, stream>>>(t1, kf_w, kf_b, t2);
  wmma_gemm_kernel<<<gUn, blk256, 0, stream>>>(t2, kun_w, kun_b, kb, LT * NPIX, DIMC, 128, 128, 1, 0);

  // v modulator
  addmask_kernel<<<gAdd, blk256, 0, stream>>>(vb, mask, modin);
  wmma_gemm_kernel<<<gSq, blk256, 0, stream>>>(modin, vsq_w, vsq_b, t1, LT * NPIX, 128, DIMC, DIMC, 1, 1);
  conv3x3_silu_kernel<<<gConv, blk256, 0, stream>>>(t1, vf_w, vf_b, t2);
  wmma_gemm_kernel<<<gUn, blk256, 0, stream>>>(t2, vun_w, vun_b, vb, LT * NPIX, DIMC, 128, 128, 1, 0);

  // pooled tokens from x, then project with Wk/Wv
  const int nPoolElems = T_FR * NPOOL * DIMC;
  pool_kernel<<<dim3((nPoolElems + 255) / 256), blk256, 0, stream>>>(x, pool_w, pool_b, pxb);
  const dim3 gemmPool(T_FR * NPOOL / 64, DIMC / 64);   // 9 x 8
  wmma_gemm_kernel<<<gemmPool, blk256, 0, stream>>>(pxb, Wk, bk, pkb, T_FR * NPOOL, DIMC, DIMC, DIMC, 0, 0);
  wmma_gemm_kernel<<<gemmPool, blk256, 0, stream>>>(pxb, Wv, bv, pvb, T_FR * NPOOL, DIMC, DIMC, DIMC, 0, 0);

  // per-window masked flags + gather tables
  flags_kernel<<<dim3(NWIN), dim3(64), 0, stream>>>(mask, flags);
  idxtab_kernel<<<dim3(1), dim3(64), 0, stream>>>(idxTab);

  // attention (masked: global over t*NK = 1600 keys; unmasked: per-frame 64x64)
  attn_global_kernel<<<dim3(16, NHEAD, NWIN), dim3(32), 0, stream>>>(
      qb, kb, vb, pkb, pvb, idxTab, flags, yb);
  attn_local_kernel<<<dim3(16, NHEAD, NWIN), dim3(32), 0, stream>>>(
      qb, kb, vb, flags, yb);

  // output projection
  wmma_gemm_kernel<<<gemmFull, blk256, 0, stream>>>(yb, Wp, bp, out, NTOK, DIMC, DIMC, DIMC, 0, 0);
}